// Group_73907797230024
// MI455X (gfx1250) — compile-verified
//
#include <hip/hip_runtime.h>
#include <hip/hip_bf16.h>

// Problem sizes (match reference)
#define NPTS     65536   // N
#define NGRP     2048    // G
#define KNN      64      // M
#define SLAB_PTS 4096    // points staged per TDM slab (48 KB)
#define NSLAB    (NPTS / SLAB_PTS)

typedef float v2f __attribute__((ext_vector_type(2)));
typedef float v8f __attribute__((ext_vector_type(8)));
typedef unsigned int u32x4 __attribute__((ext_vector_type(4)));
typedef int          i32x4v __attribute__((ext_vector_type(4)));
typedef int          i32x8v __attribute__((ext_vector_type(8)));

// ---------------------------------------------------------------------------
// Kernel 1: Farthest Point Sampling (unchanged from round 1 -- compiles with
// 256 KB LDS resident min-distance array; latency-serial so TDM won't help).
// ---------------------------------------------------------------------------
__global__ __launch_bounds__(1024) void fps_kernel(const float* __restrict__ pc,
                                                   const int* __restrict__ seedp,
                                                   float* __restrict__ center) {
  __shared__ float    mind[NPTS];     // 256 KB: running min sq-distance
  __shared__ float    s_red[32];
  __shared__ unsigned s_redi[32];
  __shared__ unsigned s_next;

  const int t = threadIdx.x;

  for (int j = t; j < NPTS; j += 1024) mind[j] = 3.402823466e38f;

  if (t == 0) {
    unsigned s = (unsigned)seedp[0];
    s += 0x9e3779b9u; s ^= s >> 16; s *= 0x85ebca6bu;
    s ^= s >> 13;     s *= 0xc2b2ae35u; s ^= s >> 16;
    s_next = s & (NPTS - 1);
  }
  __syncthreads();

  for (int i = 0; i < NGRP; ++i) {
    const unsigned last = s_next;
    const float cx = pc[3 * last + 0];
    const float cy = pc[3 * last + 1];
    const float cz = pc[3 * last + 2];
    if (t == 0) {
      center[3 * i + 0] = cx;
      center[3 * i + 1] = cy;
      center[3 * i + 2] = cz;
    }

    float    bestv = -1.0f;
    unsigned bestj = 0u;
    for (int k = 0; k < NPTS / 1024; ++k) {
      const int   j  = t + (k << 10);
      const float dx = pc[3 * j + 0] - cx;
      const float dy = pc[3 * j + 1] - cy;
      const float dz = pc[3 * j + 2] - cz;
      const float d  = dx * dx + dy * dy + dz * dz;
      const float m  = fminf(mind[j], d);
      mind[j] = m;
      if (m > bestv || (m == bestv && (unsigned)j < bestj)) { bestv = m; bestj = (unsigned)j; }
    }

    for (int off = 16; off > 0; off >>= 1) {
      const float    ov = __shfl_down(bestv, off);
      const unsigned oj = __shfl_down(bestj, off);
      if (ov > bestv || (ov == bestv && oj < bestj)) { bestv = ov; bestj = oj; }
    }
    const int wid = t >> 5;
    if ((t & 31) == 0) { s_red[wid] = bestv; s_redi[wid] = bestj; }
    __syncthreads();
    if (t < 32) {
      bestv = s_red[t];
      bestj = s_redi[t];
      for (int off = 16; off > 0; off >>= 1) {
        const float    ov = __shfl_down(bestv, off);
        const unsigned oj = __shfl_down(bestj, off);
        if (ov > bestv || (ov == bestv && oj < bestj)) { bestv = ov; bestj = oj; }
      }
      if (t == 0) s_next = bestj;
    }
    __syncthreads();
  }
}

// ---------------------------------------------------------------------------
// TDM: issue one async DMA of a 4096-point (12288-dword) slab of pc into LDS.
// D# packing per CDNA5 ISA:
//   group0: count=1 | lds_addr | global_addr[56:0] | type=2
//   group1: data_size=4B, tensor_dim0=3*NPTS, tensor_dim1=1,
//           tile_dim0=12288 dwords, tile_dim1=1, tensor_dim0_stride=3*NPTS
// Tracked by TENSORcnt (waited with s_wait_tensorcnt by the issuing wave).
// ---------------------------------------------------------------------------
__device__ __forceinline__ void tdm_issue_slab(const float* __restrict__ pc,
                                               unsigned lds_byte_off, int slab) {
  const unsigned long long ga =
      (unsigned long long)(uintptr_t)(pc + (size_t)slab * SLAB_PTS * 3);
  u32x4 g0;
  g0.x = 1u;                                              // count=1 (valid user D#)
  g0.y = lds_byte_off;                                    // lds_addr
  g0.z = (unsigned)ga;                                    // global_addr[95:64]
  g0.w = ((unsigned)(ga >> 32) & 0x01FFFFFFu) | (2u << 30); // ga[120:96] | type=2
  i32x8v g1;
  g1[0] = 0x20000;                 // workgroup_mask=0, data_size=2 (4 bytes)
  g1[1] = 0;                       // abar=0, tensor_dim0 lo16 (0x30000 -> 0)
  g1[2] = (1 << 16) | 3;           // tensor_dim0 hi16=3, tensor_dim1 lo16=1
  g1[3] = (int)(12288u << 16);     // tensor_dim1 hi16=0, tile_dim0=12288
  g1[4] = 1;                       // tile_dim1=1, tile_dim2=0
  g1[5] = 0x30000;                 // tensor_dim0_stride lo32 = 3*NPTS
  g1[6] = 0;                       // stride hi16 / dim1_stride lo16
  g1[7] = 0;
  i32x4v gz; gz[0] = 0; gz[1] = 0; gz[2] = 0; gz[3] = 0;
#if __clang_major__ >= 23
  i32x8v gz8; for (int i = 0; i < 8; ++i) gz8[i] = 0;
  __builtin_amdgcn_tensor_load_to_lds(g0, g1, gz, gz, gz8, 0);
#else
  __builtin_amdgcn_tensor_load_to_lds(g0, g1, gz, gz, 0);
#endif
}

// ---------------------------------------------------------------------------
// One streaming pass over all N points: TDM double-buffered LDS staging,
// V_WMMA_F32_16X16X4_F32 distance tiles, pass-specific LDS radix work.
//   PASS 1: 11-bit histogram   PASS 2: refine next 11 bits   PASS 3: collect
// ---------------------------------------------------------------------------
template <int PASS>
__device__ __forceinline__ void stream_pass(
    const float* __restrict__ pc, float* __restrict__ stage, unsigned stage_off,
    v2f afrag, const float* __restrict__ qsq, int t, int lane, int wave, int mb,
    unsigned* hist, const unsigned* b1, const unsigned* tp,
    const unsigned* cstrict, unsigned* ctrA, unsigned* ctrB,
    unsigned* key_l, unsigned* idx_l) {
  if (t == 0) tdm_issue_slab(pc, stage_off, 0);          // prime the pipeline
  for (int s = 0; s < NSLAB; ++s) {
    if (t == 0) __builtin_amdgcn_s_wait_tensorcnt(0);    // slab s landed in LDS
    __syncthreads();
    if (t == 0 && s + 1 < NSLAB)                         // overlap DMA of s+1
      tdm_issue_slab(pc, stage_off + (unsigned)((s + 1) & 1) * (SLAB_PTS * 12u),
                     s + 1);
    const float* buf   = stage + (size_t)(s & 1) * SLAB_PTS * 3;
    const int    nbase = s * SLAB_PTS;

    for (int tl = wave; tl < SLAB_PTS / 16; tl += 8) {
      const int   p  = (tl << 4) + (lane & 15);
      const float rx = buf[3 * p + 0];
      const float ry = buf[3 * p + 1];
      const float rz = buf[3 * p + 2];
      const float rs = rx * rx + ry * ry + rz * rz;
      const bool  lo = lane < 16;
      v2f b;
      b[0] = lo ? rx : rz;
      b[1] = lo ? ry : rs;
      v8f c;
#pragma unroll
      for (int v = 0; v < 8; ++v) c[v] = qsq[mb + v];    // C init = |q|^2
      c = __builtin_amdgcn_wmma_f32_16x16x4_f32(false, afrag, false, b,
                                                (short)0, c, false, false);
      const int n0 = nbase + (tl << 4) + (lane & 15);
#pragma unroll
      for (int v = 0; v < 8; ++v) {
        const int      mr  = mb + v;
        const unsigned key = __float_as_uint(fmaxf(c[v], 0.0f));
        if constexpr (PASS == 1) {
          atomicAdd(&hist[mr * 2048 + (key >> 21)], 1u);
        } else if constexpr (PASS == 2) {
          if ((key >> 21) == b1[mr])
            atomicAdd(&hist[mr * 2048 + ((key >> 10) & 2047u)], 1u);
        } else {
          const unsigned k22 = key >> 10;
          if (k22 < tp[mr]) {
            const unsigned pos = atomicAdd(&ctrA[mr], 1u);
            key_l[mr * KNN + pos] = key;
            idx_l[mr * KNN + pos] = (unsigned)n0;
          } else if (k22 == tp[mr]) {
            const unsigned pos = atomicAdd(&ctrB[mr], 1u);
            if (pos < KNN - cstrict[mr]) {
              const unsigned sl = cstrict[mr] + pos;
              key_l[mr * KNN + sl] = key;
              idx_l[mr * KNN + sl] = (unsigned)n0;
            }
          }
        }
      }
    }
    __syncthreads();   // buf[s&1] free before it is re-targeted at slab s+2
  }
}

// ---------------------------------------------------------------------------
// Kernel 2: kNN top-64 per center + gather + center-subtract.
// 256 threads (8 waves) per 16 centers; distances via f32 WMMA; selection via
// 2-level 11-bit LDS radix select; pc staged by TDM (double-buffered).
// LDS: 128K hist + 96K stage + 8K lists  (~238 KB of the 320 KB WGP LDS).
// ---------------------------------------------------------------------------
__global__ __launch_bounds__(256) void knn_kernel(const float* __restrict__ pc,
                                                  const float* __restrict__ center,
                                                  float* __restrict__ out_neigh) {
  __shared__ unsigned hist[16 * 2048];            // 128 KB radix histogram
  __shared__ float    stage[2 * SLAB_PTS * 3];    // 96 KB TDM double buffer
  __shared__ unsigned key_l[16 * KNN];
  __shared__ unsigned idx_l[16 * KNN];
  __shared__ unsigned b1[16], cb1[16], tp[16], cstrict[16];
  __shared__ unsigned ctrA[16], ctrB[16];
  __shared__ float    cxs[16], cys[16], czs[16], qsq[16];

  const int t    = threadIdx.x;
  const int lane = t & 31;
  const int wave = t >> 5;
  const int g0   = blockIdx.x * 16;

  // Flat LDS-aperture address truncates to the LDS byte offset (ISA 10.2).
  const unsigned stage_off = (unsigned)(uintptr_t)&stage[0];

  if (t < 16) {
    const float cx = center[3 * (g0 + t) + 0];
    const float cy = center[3 * (g0 + t) + 1];
    const float cz = center[3 * (g0 + t) + 2];
    cxs[t] = cx; cys[t] = cy; czs[t] = cz;
    qsq[t] = cx * cx + cy * cy + cz * cz;
    ctrA[t] = 0u; ctrB[t] = 0u;
  }
  for (int i = t; i < 16 * 2048; i += 256) hist[i] = 0u;
  __syncthreads();

  // Wave-uniform A fragment: row m = (-2cx, -2cy, -2cz, 1).
  const int  ma = lane & 15;
  const bool lo = lane < 16;
  v2f afrag;
  afrag[0] = lo ? (-2.0f * cxs[ma]) : (-2.0f * czs[ma]);
  afrag[1] = lo ? (-2.0f * cys[ma]) : 1.0f;
  const int mb = lo ? 0 : 8;

  // -------- pass 1: histogram of top 11 bits --------
  stream_pass<1>(pc, stage, stage_off, afrag, qsq, t, lane, wave, mb,
                 hist, b1, tp, cstrict, ctrA, ctrB, key_l, idx_l);
  if (t < 16) {
    unsigned cum = 0u, bb = 2047u, cb = 0u; bool f = false;
    for (int b = 0; b < 2048; ++b) {
      const unsigned h = hist[t * 2048 + b];
      if (!f && cum + h >= KNN) { bb = (unsigned)b; cb = cum; f = true; }
      cum += h;
    }
    b1[t] = bb; cb1[t] = cb;
  }
  __syncthreads();
  for (int i = t; i < 16 * 2048; i += 256) hist[i] = 0u;
  __syncthreads();

  // -------- pass 2: refine next 11 bits inside the bucket --------
  stream_pass<2>(pc, stage, stage_off, afrag, qsq, t, lane, wave, mb,
                 hist, b1, tp, cstrict, ctrA, ctrB, key_l, idx_l);
  if (t < 16) {
    const unsigned need = KNN - cb1[t];
    unsigned cum = 0u, bb = 2047u, cb = 0u; bool f = false;
    for (int b = 0; b < 2048; ++b) {
      const unsigned h = hist[t * 2048 + b];
      if (!f && cum + h >= need) { bb = (unsigned)b; cb = cum; f = true; }
      cum += h;
    }
    tp[t]      = (b1[t] << 11) | bb;   // 22-bit key prefix of the 64th element
    cstrict[t] = cb1[t] + cb;          // # strictly below that prefix (<= 63)
  }
  __syncthreads();

  // -------- pass 3: collect indices --------
  stream_pass<3>(pc, stage, stage_off, afrag, qsq, t, lane, wave, mb,
                 hist, b1, tp, cstrict, ctrA, ctrB, key_l, idx_l);
  __syncthreads();

  // -------- sort each center's 64 by (distance, index) ascending --------
  if (t < 16) {
    unsigned* K = &key_l[t * KNN];
    unsigned* I = &idx_l[t * KNN];
    for (int a = 1; a < KNN; ++a) {
      const unsigned kv = K[a], iv = I[a];
      int b = a - 1;
      while (b >= 0 && (K[b] > kv || (K[b] == kv && I[b] > iv))) {
        K[b + 1] = K[b]; I[b + 1] = I[b]; --b;
      }
      K[b + 1] = kv; I[b + 1] = iv;
    }
  }
  __syncthreads();

  // -------- gather + subtract center --------
  for (int e = t; e < 16 * KNN; e += 256) {
    const int      m  = e >> 6;
    const int      j  = e & (KNN - 1);
    const int      gi = g0 + m;
    const unsigned id = idx_l[m * KNN + j];
    const long     ob = (long)(gi * KNN + j) * 3;
    out_neigh[ob + 0] = pc[3 * id + 0] - cxs[m];
    out_neigh[ob + 1] = pc[3 * id + 1] - cys[m];
    out_neigh[ob + 2] = pc[3 * id + 2] - czs[m];
  }
}

// ---------------------------------------------------------------------------
extern "C" void kernel_launch(void* const* d_in, const int* in_sizes, int n_in,
                              void* d_out, int out_size, void* d_ws, size_t ws_size,
                              hipStream_t stream) {
  (void)in_sizes; (void)n_in; (void)d_ws; (void)ws_size; (void)out_size;
  const float* pc   = (const float*)d_in[0];
  const int*   seed = (const int*)d_in[1];
  float* out   = (float*)d_out;
  float* neigh = out;                                // [G*M*3]
  float* centr = out + (long)NGRP * KNN * 3;         // [G*3]

  fps_kernel<<<1, 1024, 0, stream>>>(pc, seed, centr);
  knn_kernel<<<NGRP / 16, 256, 0, stream>>>(pc, centr, neigh);
}